// SpatialTransformer_59296318488651
// MI455X (gfx1250) — compile-verified
//
#include <hip/hip_runtime.h>

// ---------------------------------------------------------------------------
// TransformerConv (graph attention) for MI455X / gfx1250, fp32 end-to-end.
// N=100000 nodes, E=800000 edges, HID=128, HEADS=4, HD=32.
//
// v2: projections via V_WMMA_F32_16X16X4_F32 (unchanged, codegen-verified).
//     Edge aggregation switched from 102M global f32 atomics to a CSR
//     build (int atomics only) + gather-based per-node softmax/weighted-sum
//     with one wave32 per (node, head). q/k/v (153 MB) stay L2-resident.
// ---------------------------------------------------------------------------

#define HID 128
#define HEADS 4
#define HD 32

typedef float v2f __attribute__((ext_vector_type(2)));
typedef float v8f __attribute__((ext_vector_type(8)));

#if defined(__HIP_DEVICE_COMPILE__) && !__has_builtin(__builtin_amdgcn_wmma_f32_16x16x4_f32)
#error "missing __builtin_amdgcn_wmma_f32_16x16x4_f32 on device"
#endif

// ---------------------------------------------------------------------------
// K0: zero an int array (degree counters)
// ---------------------------------------------------------------------------
__global__ void zero_int_kernel(int* __restrict__ p, int n) {
  int t = blockIdx.x * blockDim.x + threadIdx.x;
  if (t < n) p[t] = 0;
}

// ---------------------------------------------------------------------------
// K1: OUT[r, :] = X[r, :] @ W + b   via V_WMMA_F32_16X16X4_F32
// block = 256 threads = 8 waves; each wave computes a 16x128 row block.
// W (128x128 f32 = 64 KB) staged in LDS.
// ---------------------------------------------------------------------------
__global__ __launch_bounds__(256) void proj_kernel(
    const float* __restrict__ X, const float* __restrict__ W,
    const float* __restrict__ bias, float* __restrict__ OUT, int nrows) {
#if defined(__HIP_DEVICE_COMPILE__)
  __shared__ float sW[HID * HID];  // 64 KB

  {
    const float4* Wv = (const float4*)W;
    float4* sWv = (float4*)sW;
#pragma unroll
    for (int i = 0; i < 16; ++i) sWv[threadIdx.x + i * 256] = Wv[threadIdx.x + i * 256];
  }
  __syncthreads();

  const int wave = threadIdx.x >> 5;  // 0..7
  const int lane = threadIdx.x & 31;
  const int row0 = (blockIdx.x * 8 + wave) * 16;
  if (row0 >= nrows) return;  // wave-uniform: EXEC stays all-1s for WMMA

  const int m = lane & 15;      // A row (M) / C col (N)
  const int khalf = lane >> 4;  // 0: K={0,1}, 1: K={2,3}

  v8f c[8];
#pragma unroll
  for (int t = 0; t < 8; ++t) c[t] = v8f{0.f, 0.f, 0.f, 0.f, 0.f, 0.f, 0.f, 0.f};

  for (int k4 = 0; k4 < HID; k4 += 4) {
    v2f a = *(const v2f*)(X + (size_t)(row0 + m) * HID + k4 + khalf * 2);
#pragma unroll
    for (int t = 0; t < 8; ++t) {
      const float* wp = sW + (k4 + khalf * 2) * HID + t * 16 + m;
      v2f b;
      b.x = wp[0];
      b.y = wp[HID];
      c[t] = __builtin_amdgcn_wmma_f32_16x16x4_f32(
          false, a, false, b, (short)0, c[t], false, false);
    }
  }

#pragma unroll
  for (int t = 0; t < 8; ++t) {
    const int col = t * 16 + m;
    const float bv = bias[col];
#pragma unroll
    for (int r = 0; r < 8; ++r) {
      const int row = row0 + r + khalf * 8;
      OUT[(size_t)row * HID + col] = c[t][r] + bv;
    }
  }
#endif
}

// ---------------------------------------------------------------------------
// K2: per (edge, head) attention logit -> alpha[e*4+h]
// ---------------------------------------------------------------------------
__global__ void edge_logits_kernel(const long long* __restrict__ ei,
                                   const float* __restrict__ q,
                                   const float* __restrict__ k,
                                   float* __restrict__ alpha, int E) {
  int t = blockIdx.x * blockDim.x + threadIdx.x;
  if (t >= E * HEADS) return;
  const int e = t >> 2, h = t & 3;
  const int src = (int)ei[e];
  const int dst = (int)ei[E + e];
  const float4* qp = (const float4*)(q + (size_t)dst * HID + h * HD);
  const float4* kp = (const float4*)(k + (size_t)src * HID + h * HD);
  float s = 0.0f;
#pragma unroll
  for (int i = 0; i < HD / 4; ++i) {
    float4 a = qp[i], b = kp[i];
    s += a.x * b.x + a.y * b.y + a.z * b.z + a.w * b.w;
  }
  alpha[t] = s * 0.17677669529663687f;  // 1/sqrt(32)
}

// ---------------------------------------------------------------------------
// K3: per-destination degree histogram
// ---------------------------------------------------------------------------
__global__ void count_kernel(const long long* __restrict__ ei,
                             int* __restrict__ deg, int E) {
  int e = blockIdx.x * blockDim.x + threadIdx.x;
  if (e >= E) return;
  atomicAdd(deg + (int)ei[E + e], 1);
}

// ---------------------------------------------------------------------------
// K4: single-block exclusive prefix scan: deg[0..n) -> rowptr[0..n]
// 1024 threads, chunked Hillis-Steele scan with running carry.
// ---------------------------------------------------------------------------
__global__ __launch_bounds__(1024) void scan_kernel(const int* __restrict__ deg,
                                                    int* __restrict__ rowptr, int n) {
  __shared__ int sm[1024];
  __shared__ int carry;
  const int tid = threadIdx.x;
  if (tid == 0) carry = 0;
  __syncthreads();
  for (int base = 0; base < n; base += 1024) {
    const int i = base + tid;
    const int v = (i < n) ? deg[i] : 0;
    sm[tid] = v;
    __syncthreads();
#pragma unroll
    for (int off = 1; off < 1024; off <<= 1) {
      const int add = (tid >= off) ? sm[tid - off] : 0;
      __syncthreads();
      sm[tid] += add;
      __syncthreads();
    }
    const int incl = sm[tid];
    const int c = carry;
    if (i < n) rowptr[i] = c + incl - v;
    const int total = sm[1023];
    __syncthreads();
    if (tid == 0) carry = c + total;
    __syncthreads();
  }
  if (tid == 0) rowptr[n] = carry;
}

// ---------------------------------------------------------------------------
// K5: cursor = rowptr copy
// ---------------------------------------------------------------------------
__global__ void copy_int_kernel(const int* __restrict__ a, int* __restrict__ b, int n) {
  int t = blockIdx.x * blockDim.x + threadIdx.x;
  if (t < n) b[t] = a[t];
}

// ---------------------------------------------------------------------------
// K6: CSR fill — slot per edge via atomic cursor; store edge id + src node
// ---------------------------------------------------------------------------
__global__ void fill_kernel(const long long* __restrict__ ei,
                            int* __restrict__ cursor,
                            int* __restrict__ eid, int* __restrict__ esrc, int E) {
  int e = blockIdx.x * blockDim.x + threadIdx.x;
  if (e >= E) return;
  const int src = (int)ei[e];
  const int dst = (int)ei[E + e];
  const int pos = atomicAdd(cursor + dst, 1);
  eid[pos] = e;
  esrc[pos] = src;
}

// ---------------------------------------------------------------------------
// K7: gather aggregation — one wave32 per (node, head); lane = head dim.
// Pass 1: segment max of logits. Pass 2: s = sum(exp), accd = sum(exp * v).
// acc (skip connection) updated in place: acc += accd / (s + 1e-16).
// ---------------------------------------------------------------------------
__global__ __launch_bounds__(256) void aggregate_kernel(
    const int* __restrict__ rowptr, const int* __restrict__ eid,
    const int* __restrict__ esrc, const float* __restrict__ alpha,
    const float* __restrict__ v, float* __restrict__ acc, int N) {
  const int wid = (blockIdx.x * blockDim.x + threadIdx.x) >> 5;
  const int lane = threadIdx.x & 31;
  if (wid >= N * HEADS) return;
  const int node = wid >> 2;
  const int h = wid & 3;

  const int beg = rowptr[node];
  const int end = rowptr[node + 1];

  // pass 1: max logit (all lanes redundantly compute the same value)
  float m = -3.402823466e+38f;
  for (int j = beg; j < end; ++j) {
    const float a = alpha[(size_t)eid[j] * HEADS + h];
    m = fmaxf(m, a);
  }

  // pass 2: exp-sum and weighted value accumulation
  float s = 0.0f;
  float accd = 0.0f;
  for (int j = beg; j < end; ++j) {
    const float a = alpha[(size_t)eid[j] * HEADS + h];
    const float ex = __expf(a - m);
    s += ex;
    accd += ex * v[(size_t)esrc[j] * HID + h * HD + lane];
  }

  const size_t o = (size_t)node * HID + h * HD + lane;
  acc[o] = acc[o] + accd / (s + 1e-16f);
}

// ---------------------------------------------------------------------------
// K8: LayerNorm — one wave32 per node, 4 elements per lane
// ---------------------------------------------------------------------------
__global__ __launch_bounds__(256) void layernorm_kernel(
    const float* __restrict__ acc, const float* __restrict__ gamma,
    const float* __restrict__ beta, float* __restrict__ out, int n) {
  const int node = (blockIdx.x * blockDim.x + threadIdx.x) >> 5;
  const int lane = threadIdx.x & 31;
  if (node >= n) return;
  const float4 x = ((const float4*)(acc + (size_t)node * HID))[lane];
  float s = x.x + x.y + x.z + x.w;
  float s2 = x.x * x.x + x.y * x.y + x.z * x.z + x.w * x.w;
#pragma unroll
  for (int off = 16; off > 0; off >>= 1) {
    s += __shfl_xor(s, off, 32);
    s2 += __shfl_xor(s2, off, 32);
  }
  const float mu = s * (1.0f / HID);
  const float var = s2 * (1.0f / HID) - mu * mu;
  const float rstd = rsqrtf(var + 1e-5f);
  const float4 g = ((const float4*)gamma)[lane];
  const float4 b = ((const float4*)beta)[lane];
  float4 o;
  o.x = (x.x - mu) * rstd * g.x + b.x;
  o.y = (x.y - mu) * rstd * g.y + b.y;
  o.z = (x.z - mu) * rstd * g.z + b.z;
  o.w = (x.w - mu) * rstd * g.w + b.w;
  ((float4*)(out + (size_t)node * HID))[lane] = o;
}

// ---------------------------------------------------------------------------
// Launch
// ---------------------------------------------------------------------------
extern "C" void kernel_launch(void* const* d_in, const int* in_sizes, int n_in,
                              void* d_out, int out_size, void* d_ws, size_t ws_size,
                              hipStream_t stream) {
  (void)n_in; (void)out_size; (void)ws_size;

  const float* x = (const float*)d_in[0];
  const long long* ei = (const long long*)d_in[1];
  const float* Wq = (const float*)d_in[2];
  const float* bq = (const float*)d_in[3];
  const float* Wk = (const float*)d_in[4];
  const float* bk = (const float*)d_in[5];
  const float* Wv = (const float*)d_in[6];
  const float* bv = (const float*)d_in[7];
  const float* Ws = (const float*)d_in[8];
  const float* bs = (const float*)d_in[9];
  const float* gamma = (const float*)d_in[10];
  const float* beta = (const float*)d_in[11];
  float* out = (float*)d_out;

  const int N = in_sizes[0] / HID;  // 100000
  const int E = in_sizes[1] / 2;    // 800000

  // workspace layout
  float* wsf = (float*)d_ws;
  float* q = wsf;                            // N*128
  float* kbuf = q + (size_t)N * HID;         // N*128
  float* vbuf = kbuf + (size_t)N * HID;      // N*128
  float* acc = vbuf + (size_t)N * HID;       // N*128 (skip + aggregation)
  float* alpha = acc + (size_t)N * HID;      // E*4
  int* deg = (int*)(alpha + (size_t)E * HEADS);  // N
  int* rowptr = deg + N;                         // N+1
  int* cursor = rowptr + (N + 1);                // N
  int* eid = cursor + N;                         // E
  int* esrc = eid + E;                           // E

  // K0: zero degrees
  zero_int_kernel<<<(N + 255) / 256, 256, 0, stream>>>(deg, N);

  // K1: projections (WMMA f32): q, k, v and skip (acc = x@Ws + bs)
  const int pb = (N + 127) / 128;
  proj_kernel<<<pb, 256, 0, stream>>>(x, Wq, bq, q, N);
  proj_kernel<<<pb, 256, 0, stream>>>(x, Wk, bk, kbuf, N);
  proj_kernel<<<pb, 256, 0, stream>>>(x, Wv, bv, vbuf, N);
  proj_kernel<<<pb, 256, 0, stream>>>(x, Ws, bs, acc, N);

  // K2: attention logits
  edge_logits_kernel<<<(E * HEADS + 255) / 256, 256, 0, stream>>>(ei, q, kbuf, alpha, E);

  // K3..K6: CSR build
  count_kernel<<<(E + 255) / 256, 256, 0, stream>>>(ei, deg, E);
  scan_kernel<<<1, 1024, 0, stream>>>(deg, rowptr, N);
  copy_int_kernel<<<(N + 255) / 256, 256, 0, stream>>>(rowptr, cursor, N);
  fill_kernel<<<(E + 255) / 256, 256, 0, stream>>>(ei, cursor, eid, esrc, E);

  // K7: per-(node, head) softmax + weighted gather into acc
  aggregate_kernel<<<((size_t)N * HEADS * 32 + 255) / 256, 256, 0, stream>>>(
      rowptr, eid, esrc, alpha, vbuf, acc, N);

  // K8: layernorm -> output
  layernorm_kernel<<<((size_t)N * 32 + 255) / 256, 256, 0, stream>>>(acc, gamma, beta, out, N);
}